// PtModule_76166950027905
// MI455X (gfx1250) — compile-verified
//
#include <hip/hip_runtime.h>
#include <hip/hip_bf16.h>

// out[i, j] = (2*x[i, j] + 3) / (i + 1), x: [8192, 8192] f32.
// HBM-bound streaming kernel (512 MB traffic, ~22 us floor at 23.3 TB/s).
// Round 2: amortize the per-wave SALU dispatch preamble by moving 4 float4s
// per thread (128 B traffic per lane, 4 KB per wave) with a blockDim-strided
// unroll so each wave's accesses stay fully coalesced. All 4 non-temporal
// 128-bit loads issue before any use (MLP), then pk_fma + IEEE div + NT store
// per chunk. NT hints on both directions: 512 MB working set >> 192 MB L2,
// touched exactly once.

typedef float v4f __attribute__((ext_vector_type(4)));

#define N_COLS        8192
#define VEC4_PER_ROW  (N_COLS / 4)          // 2048 float4 per row
#define BLOCK_THREADS 256                   // 8 wave32s
#define UNROLL        4                     // float4s per thread
#define VEC4_PER_BLK  (BLOCK_THREADS * UNROLL)  // 1024 float4 = half row

__global__ __launch_bounds__(BLOCK_THREADS)
void rowdiv_stream_kernel(const float* __restrict__ x, float* __restrict__ out) {
    const int row = (int)blockIdx.y;                  // uniform per block
    const int b0  = (int)(blockIdx.x * VEC4_PER_BLK + threadIdx.x);

    const long long rowbase = (long long)row * VEC4_PER_ROW;

    const v4f* __restrict__ xv = (const v4f*)x;
    v4f*       __restrict__ ov = (v4f*)out;

    // Issue all 4 NT loads up front (blockDim-strided -> coalesced per wave).
    v4f v[UNROLL];
#pragma unroll
    for (int k = 0; k < UNROLL; ++k) {
        v[k] = __builtin_nontemporal_load(xv + rowbase + b0 + k * BLOCK_THREADS);
    }

    const float d = (float)(row + 1);   // scalar-uniform divisor

#pragma unroll
    for (int k = 0; k < UNROLL; ++k) {
        v4f r;
        r.x = __builtin_fmaf(v[k].x, 2.0f, 3.0f) / d;
        r.y = __builtin_fmaf(v[k].y, 2.0f, 3.0f) / d;
        r.z = __builtin_fmaf(v[k].z, 2.0f, 3.0f) / d;
        r.w = __builtin_fmaf(v[k].w, 2.0f, 3.0f) / d;
        __builtin_nontemporal_store(r, ov + rowbase + b0 + k * BLOCK_THREADS);
    }
}

extern "C" void kernel_launch(void* const* d_in, const int* in_sizes, int n_in,
                              void* d_out, int out_size, void* d_ws, size_t ws_size,
                              hipStream_t stream) {
    (void)in_sizes; (void)n_in; (void)d_ws; (void)ws_size;

    const float* x  = (const float*)d_in[0];
    float*      out = (float*)d_out;

    const int n_rows = out_size / N_COLS;             // 8192

    dim3 grid(VEC4_PER_ROW / VEC4_PER_BLK, n_rows, 1); // (2, 8192)
    dim3 block(BLOCK_THREADS, 1, 1);

    rowdiv_stream_kernel<<<grid, block, 0, stream>>>(x, out);
}